// BasicNCA2DFast_60490319397261
// MI455X (gfx1250) — compile-verified
//
#include <hip/hip_runtime.h>

// NCA forward, 10 steps. Internal layout: NHWC f32 (matches input x and output).
// Per step: K1 conv+fc0+batch-stat atomics, K2 finalize scale/shift, K3
// conv+fc0 (recompute) + normalize + relu + fc1 (WMMA) + masked residual.
// conv_b / fc0_b are per-hidden-channel constants through fc0 and cancel
// exactly under batchnorm mean subtraction, so they are skipped.
// Workspace: 2 state buffers (2 * 32 MiB) + 512 floats of stats.
// Weight staging into padded LDS uses the Tensor Data Mover (D# pad feature
// implements the stride-33 bank padding in the DMA engine).

namespace {
constexpr int kB   = 8;
constexpr int kH   = 256;
constexpr int kW   = 256;
constexpr int kC   = 16;
constexpr int kHid = 128;
constexpr int kTW  = 16;            // tile width  (pixels)
constexpr int kTH  = 8;             // tile height (pixels)
constexpr int kPix = kTW * kTH;     // 128 pixels per workgroup
constexpr int kHaloW = kTW + 6;     // 22
constexpr int kHaloH = kTH + 6;     // 14
constexpr int kCStr  = 17;          // 16-elem rows padded (gcd(17,64)=1)
constexpr int kW0Str = 33;          // 32-elem rows padded (gcd(33,64)=1)
constexpr long kNPix = (long)kB * kH * kW;  // 524288
}

typedef __attribute__((ext_vector_type(2))) float v2f;
typedef __attribute__((ext_vector_type(8))) float v8f;

#if defined(__HIP_DEVICE_COMPILE__) && \
    __has_builtin(__builtin_amdgcn_tensor_load_to_lds)
#define USE_TDM 1
typedef __attribute__((ext_vector_type(4))) unsigned int v4u;
typedef __attribute__((ext_vector_type(8))) int v8i;
typedef __attribute__((ext_vector_type(4))) int v4i;

// 2D TDM load of 4-byte elements: tile (tile0 x tile1), row stride t0stride
// elements, optional LDS padding of +1 DWORD every 32 DWORDs (stride 32->33).
// This toolchain exposes the 6-arg builtin:
//   (uint32x4 g0, int32x8 g1, int32x4 g2, int32x4 g3, int32x8 g4, i32 cpol)
__device__ __forceinline__ void tdm_load_2d(unsigned lds_off, const void* gptr,
                                            unsigned tile0, unsigned tile1,
                                            unsigned t0stride, int pad33) {
  unsigned long long ga = (unsigned long long)(size_t)gptr;
  v4u g0;
  g0[0] = 1u;                                     // count=1, user descriptor
  g0[1] = lds_off;                                // lds_addr (bytes)
  g0[2] = (unsigned)(ga & 0xffffffffu);           // global_addr[31:0]
  g0[3] = (unsigned)((ga >> 32) & 0x01ffffffu) | (2u << 30);  // type=2
  unsigned d0 = (2u << 16);                       // data_size = 4 bytes
  if (pad33) d0 |= (1u << 20) | (4u << 22);       // pad 1 DW per 32 DW
  unsigned dim0 = tile0, dim1 = tile1;            // dense tensor == tile
  v8i g1;
  g1[0] = (int)d0;
  g1[1] = (int)((dim0 & 0xffffu) << 16);          // [63:48]=dim0 lo16
  g1[2] = (int)((dim0 >> 16) | ((dim1 & 0xffffu) << 16));
  g1[3] = (int)((dim1 >> 16) | (tile0 << 16));    // [127:112]=tile_dim0
  g1[4] = (int)(tile1 & 0xffffu);                 // tile_dim1; tile_dim2=0
  g1[5] = (int)t0stride;                          // tensor_dim0_stride lo
  g1[6] = 0;
  g1[7] = 0;
  v4i gz;
  gz[0] = gz[1] = gz[2] = gz[3] = 0;
  v8i gz8;
#pragma unroll
  for (int i = 0; i < 8; ++i) gz8[i] = 0;
  __builtin_amdgcn_tensor_load_to_lds(g0, g1, gz, gz, gz8, 0);
}
#else
#define USE_TDM 0
#endif

__device__ __forceinline__ int reflect_idx(int i, int n) {
  i = (i < 0) ? -i : i;
  return (i >= n) ? (2 * n - 2 - i) : i;
}

__device__ __forceinline__ v8f v8f_zero() {
  v8f z;
#pragma unroll
  for (int i = 0; i < 8; ++i) z[i] = 0.0f;
  return z;
}

// Shared prologue: stage halo tile (reflect pad) + weights, depthwise 7x7 conv
// into dbuf, then fc0 (524288x32 @ 32x128 per-pixel GEMM) via f32 WMMA.
// Wave w owns pixels [16w, 16w+16): acc[nt] is the 16x16 f32 C tile for
// hidden channels [16*nt, 16*nt+16).
__device__ __forceinline__ void conv_fc0(
    const float* __restrict__ src, const float* __restrict__ convw,
    const float* __restrict__ fc0w,
    float* halo, float* dbuf, float* w0, float* wc,
    int b, int th0, int tw0, v8f acc[8]) {
  const int tid = threadIdx.x;

#if USE_TDM
  if (tid < 32) {  // wave 0 issues the weight DMAs (TDM is per-wave)
    tdm_load_2d((unsigned)(size_t)(void*)w0, fc0w, 32u, 128u, 32u, 1);
    tdm_load_2d((unsigned)(size_t)(void*)wc, convw, 784u, 1u, 784u, 0);
  }
#else
  for (int idx = tid; idx < kHid * 2 * kC; idx += 256) {
    int n = idx >> 5, k = idx & 31;
    w0[n * kW0Str + k] = fc0w[idx];
  }
  for (int idx = tid; idx < kC * 49; idx += 256) wc[idx] = convw[idx];
#endif

  // ---- halo load with reflect padding: (14 x 22 x 16) ----
  for (int idx = tid; idx < kHaloH * kHaloW * kC; idx += 256) {
    int ch = idx & 15;
    int r  = idx >> 4;
    int hc = r % kHaloW;
    int hr = r / kHaloW;
    int gh = reflect_idx(th0 + hr - 3, kH);
    int gw = reflect_idx(tw0 + hc - 3, kW);
    halo[(hr * kHaloW + hc) * kCStr + ch] =
        src[(((long)b * kH + gh) * kW + gw) * kC + ch];
  }

#if USE_TDM
  if (tid < 32) __builtin_amdgcn_s_wait_tensorcnt(0);
#endif
  __syncthreads();

  // ---- depthwise 7x7 conv: thread = (pixel, 8-channel group) ----
  {
    int p = tid & (kPix - 1);
    int row = p >> 4, col = p & 15;
    int chb = (tid >> 7) * 8;
#pragma unroll
    for (int cc = 0; cc < 8; ++cc) {
      int ch = chb + cc;
      const float* wr = &wc[ch * 49];
      float a = 0.0f;
#pragma unroll
      for (int i = 0; i < 7; ++i)
#pragma unroll
        for (int j = 0; j < 7; ++j)
          a = fmaf(halo[((row + i) * kHaloW + (col + j)) * kCStr + ch],
                   wr[i * 7 + j], a);
      dbuf[p * kCStr + ch] = a;
    }
  }
  __syncthreads();

  // ---- fc0 via V_WMMA_F32_16X16X4_F32 ----
  const int wave = tid >> 5;
  const int lane = tid & 31;
  const int ml   = lane & 15;
  const int half = lane >> 4;
  const int pA   = wave * 16 + ml;              // A-row pixel for this lane
  const int rowA = pA >> 4, colA = pA & 15;
  const int haloBase = ((rowA + 3) * kHaloW + (colA + 3)) * kCStr;

#pragma unroll
  for (int nt = 0; nt < 8; ++nt) acc[nt] = v8f_zero();

  // Preload all A fragments (16 VGPRs) so the k-loop only loads B.
  v2f afr[8];
#pragma unroll
  for (int kt = 0; kt < 8; ++kt) {
    const int k0 = kt * 4 + 2 * half;           // A layout: K = 2*half + v
    afr[kt].x = (k0 < 16) ? halo[haloBase + k0] : dbuf[pA * kCStr + (k0 - 16)];
    afr[kt].y = (k0 + 1 < 16) ? halo[haloBase + k0 + 1]
                              : dbuf[pA * kCStr + (k0 + 1 - 16)];
  }

#pragma unroll
  for (int kt = 0; kt < 8; ++kt) {
    const int k0 = kt * 4 + 2 * half;
    // Batch the 8 B-fragment loads, then issue 8 independent WMMAs:
    // one s_wait_dscnt covers the whole group instead of one per WMMA.
    v2f bf[8];
#pragma unroll
    for (int nt = 0; nt < 8; ++nt) {
      const int n = nt * 16 + ml;               // B layout: lane%16 = N
      bf[nt].x = w0[n * kW0Str + k0];
      bf[nt].y = w0[n * kW0Str + k0 + 1];
    }
#pragma unroll
    for (int nt = 0; nt < 8; ++nt)
      acc[nt] = __builtin_amdgcn_wmma_f32_16x16x4_f32(
          false, afr[kt], false, bf[nt], (short)0, acc[nt], false, false);
  }
}

__global__ void __launch_bounds__(256)
nca_init(float* stats) {
  if (threadIdx.x < 2 * kHid) stats[threadIdx.x] = 0.0f;
}

__global__ void __launch_bounds__(256)
nca_stats(const float* __restrict__ src, const float* __restrict__ convw,
          const float* __restrict__ fc0w, float* __restrict__ stats) {
  __shared__ float halo[kHaloH * kHaloW * kCStr];
  __shared__ float dbuf[kPix * kCStr];
  __shared__ float w0[kHid * kW0Str];
  __shared__ float wc[kC * 49];
  __shared__ float sred[2 * kHid];

  const int tid = threadIdx.x;
  if (tid < 2 * kHid) sred[tid] = 0.0f;

  v8f acc[8];
  conv_fc0(src, convw, fc0w, halo, dbuf, w0, wc,
           blockIdx.z, blockIdx.y * kTH, blockIdx.x * kTW, acc);

  // C layout: element (M = r + 8*half, N = lane%16) = acc[nt][r].
  const int ml = tid & 15;
#pragma unroll
  for (int nt = 0; nt < 8; ++nt) {
    float s = 0.0f, s2 = 0.0f;
#pragma unroll
    for (int r = 0; r < 8; ++r) {
      float v = acc[nt][r];
      s += v;
      s2 = fmaf(v, v, s2);
    }
    const int n = nt * 16 + ml;
    atomicAdd(&sred[n], s);
    atomicAdd(&sred[kHid + n], s2);
  }
  __syncthreads();
  if (tid < 2 * kHid) atomicAdd(&stats[tid], sred[tid]);
}

__global__ void __launch_bounds__(128)
nca_finalize(float* stats, const float* __restrict__ gamma,
             const float* __restrict__ beta, float* __restrict__ sc,
             float* __restrict__ sh) {
  const int n = threadIdx.x;
  if (n < kHid) {
    const float inv  = 1.0f / (float)kNPix;
    const float mean = stats[n] * inv;
    const float var  = fmaf(-mean, mean, stats[kHid + n] * inv);
    const float rstd = rsqrtf(var + 1e-5f);
    const float scale = rstd * gamma[n];
    sc[n] = scale;
    sh[n] = fmaf(-mean, scale, beta[n]);
    stats[n] = 0.0f;           // reset accumulators for the next step
    stats[kHid + n] = 0.0f;
  }
}

__global__ void __launch_bounds__(256)
nca_update(const float* __restrict__ src, const float* __restrict__ convw,
           const float* __restrict__ fc0w, const float* __restrict__ fc1w,
           const float* __restrict__ sc, const float* __restrict__ sh,
           const unsigned char* __restrict__ mstep, float* __restrict__ dst) {
  __shared__ float halo[kHaloH * kHaloW * kCStr];
  __shared__ float dbuf[kPix * kCStr];   // conv out; reused as transpose scratch
  __shared__ float w0[kHid * kW0Str];
  __shared__ float wc[kC * 49];
  __shared__ float w1[kHid * kCStr];     // fc1_w^T padded: [o][k], k in 0..16

  const int tid = threadIdx.x;
  const int b   = blockIdx.z;
  const int th0 = blockIdx.y * kTH;
  const int tw0 = blockIdx.x * kTW;

  // fc1 weights, transposed + zero-padded 15 -> 16 output channels
  for (int idx = tid; idx < kHid * 16; idx += 256) {
    int o = idx >> 4, kk = idx & 15;
    w1[o * kCStr + kk] = (kk < 15) ? fc1w[kk * kHid + o] : 0.0f;
  }

  v8f acc[8];
  conv_fc0(src, convw, fc0w, halo, dbuf, w0, wc, b, th0, tw0, acc);

  const int wave = tid >> 5;
  const int lane = tid & 31;
  const int ml   = lane & 15;
  const int half = lane >> 4;

  // normalize + relu in place (per hidden channel affine)
#pragma unroll
  for (int nt = 0; nt < 8; ++nt) {
    const int n = nt * 16 + ml;
    const float scl = sc[n], shf = sh[n];
#pragma unroll
    for (int r = 0; r < 8; ++r) {
      float v = fmaf(acc[nt][r], scl, shf);
      acc[nt][r] = v > 0.0f ? v : 0.0f;
    }
  }
  __syncthreads();  // dbuf conv data no longer needed by any wave

  // fc1: delta(16 px x 16 ch) = h(16 x 128) @ fc1_w^T(128 x 16).
  // Per nt: transpose the C-layout 16x16 h tile through wave-private LDS
  // (reuse dbuf rows [16*wave, 16*wave+16)), then 4 x K=4 WMMA steps with
  // batched fragment loads.
  float* tsc = &dbuf[wave * 16 * kCStr];
  v8f dacc = v8f_zero();
#pragma unroll
  for (int nt = 0; nt < 8; ++nt) {
#pragma unroll
    for (int r = 0; r < 8; ++r)
      tsc[(r + 8 * half) * kCStr + ml] = acc[nt][r];  // (m, n_local)
    v2f a2[4], b2[4];
#pragma unroll
    for (int kt2 = 0; kt2 < 4; ++kt2) {
      const int ol = kt2 * 4 + 2 * half;   // local o within this 16-chunk
      a2[kt2].x = tsc[ml * kCStr + ol];    // A: row = pixel, K = o
      a2[kt2].y = tsc[ml * kCStr + ol + 1];
      const int og = nt * 16 + ol;
      b2[kt2].x = w1[og * kCStr + ml];     // B: K = o, N = out channel k
      b2[kt2].y = w1[(og + 1) * kCStr + ml];
    }
#pragma unroll
    for (int kt2 = 0; kt2 < 4; ++kt2)
      dacc = __builtin_amdgcn_wmma_f32_16x16x4_f32(
          false, a2[kt2], false, b2[kt2], (short)0, dacc, false, false);
  }

  // masked residual update; lane's N (= ml) is output channel kk.
  // kk in [0,15) -> state channel kk+1 ; kk == 15 lane rewrites channel 0.
  const int kk = ml;
  const int ch = (kk < 15) ? (kk + 1) : 0;
#pragma unroll
  for (int r = 0; r < 8; ++r) {
    const int p   = wave * 16 + r + 8 * half;
    const int row = p >> 4, col = p & 15;
    const int gh = th0 + row, gw = tw0 + col;
    const float mv =
        mstep[((long)b * kH + gh) * kW + gw] ? 1.0f : 0.0f;
    const float oldv = halo[((row + 3) * kHaloW + (col + 3)) * kCStr + ch];
    const float nv = (kk < 15) ? fmaf(dacc[r], mv, oldv) : oldv;
    dst[(((long)b * kH + gh) * kW + gw) * kC + ch] = nv;
  }
}

extern "C" void kernel_launch(void* const* d_in, const int* in_sizes, int n_in,
                              void* d_out, int out_size, void* d_ws,
                              size_t ws_size, hipStream_t stream) {
  const float* x            = (const float*)d_in[0];
  const unsigned char* mks  = (const unsigned char*)d_in[1];  // bool masks
  const float* convw        = (const float*)d_in[2];
  // d_in[3] = conv_b, d_in[5] = fc0_b: exactly cancelled by batchnorm mean.
  const float* fc0w         = (const float*)d_in[4];
  const float* fc1w         = (const float*)d_in[6];
  const float* gamma        = (const float*)d_in[7];
  const float* beta         = (const float*)d_in[8];
  float* out                = (float*)d_out;

  const size_t stateBytes = (size_t)kNPix * kC * sizeof(float);  // 32 MiB
  char* ws    = (char*)d_ws;
  float* S0   = (float*)ws;
  float* S1   = (float*)(ws + stateBytes);
  float* stats = (float*)(ws + 2 * stateBytes);  // sum[128] ++ sumsq[128]
  float* scb   = stats + 2 * kHid;
  float* shb   = scb + kHid;

  dim3 grid(kW / kTW, kH / kTH, kB);  // (16, 32, 8)
  nca_init<<<1, 256, 0, stream>>>(stats);

  const float* src = x;
  for (int s = 0; s < 10; ++s) {
    float* dst = (s == 9) ? out : ((s & 1) ? S1 : S0);
    nca_stats<<<grid, 256, 0, stream>>>(src, convw, fc0w, stats);
    nca_finalize<<<1, 128, 0, stream>>>(stats, gamma, beta, scb, shb);
    nca_update<<<grid, 256, 0, stream>>>(
        src, convw, fc0w, fc1w, scb, shb,
        mks + (size_t)s * kB * kH * kW, dst);
    src = dst;
  }
}